// RNNModel_33887291965979
// MI455X (gfx1250) — compile-verified
//
#include <hip/hip_runtime.h>
#include <hip/hip_bf16.h>

typedef __attribute__((ext_vector_type(2))) float v2f;
typedef __attribute__((ext_vector_type(8))) float v8f;
typedef __attribute__((ext_vector_type(4))) int   v4i;

#define WMMA_F32(a, b, c) \
  __builtin_amdgcn_wmma_f32_16x16x4_f32(false, (a), false, (b), (short)0, (c), false, false)

static constexpr int kB = 64;      // batch
static constexpr int kT = 512;     // seq len
static constexpr int kE = 256;     // embedding dim
static constexpr int kH = 512;     // hidden dim
static constexpr int kC = 64;      // classifier hidden
static constexpr int kO = 18;      // output classes
static constexpr int kM = kB * kT; // 32768 flattened rows

// ---- async global->LDS copy (CDNA5 GLOBAL_LOAD_ASYNC_TO_LDS_B128) ----------
__device__ __forceinline__ void cp_async16(void* lds_dst, const void* gsrc) {
#if __has_builtin(__builtin_amdgcn_global_load_async_to_lds_b128)
  __builtin_amdgcn_global_load_async_to_lds_b128(
      (__attribute__((address_space(1))) v4i*)(uintptr_t)gsrc,
      (__attribute__((address_space(3))) v4i*)(unsigned)(uintptr_t)lds_dst,
      0, 0);
#else
  *(float4*)lds_dst = *(const float4*)gsrc;
#endif
}
__device__ __forceinline__ void cp_async_wait() {
#if __has_builtin(__builtin_amdgcn_global_load_async_to_lds_b128)
#if __has_builtin(__builtin_amdgcn_s_wait_asynccnt)
  __builtin_amdgcn_s_wait_asynccnt(0);
#else
  asm volatile("s_wait_asynccnt 0x0" ::: "memory");
#endif
#endif
}

// ---------------------------------------------------------------------------
// Kernel 1: xg[m, n] = emb_table[x[m], :] @ Wih[n, :] + bih[n] + bhh[n]
// M = 32768, K = 256, N = 512.  64x64 tiles, 256 threads (8 waves).
// ---------------------------------------------------------------------------
__global__ __launch_bounds__(256) void xg_kernel(
    const int* __restrict__ x, const float* __restrict__ emb,
    const float* __restrict__ Wih, const float* __restrict__ bih,
    const float* __restrict__ bhh, float* __restrict__ XG)
{
  __shared__ float As[64][68];    // [row][k]   gathered embeddings
  __shared__ float Bs[32][160];   // paired Wih^T: [k/2][2n + (k&1)]
  __shared__ int   toks[64];

  const int tid = threadIdx.x;
  const int M0 = blockIdx.x * 64;
  const int N0 = blockIdx.y * 64;
  if (tid < 64) toks[tid] = x[M0 + tid];

  const int w = tid >> 5, lane = tid & 31;
  const int ml = lane & 15, hi = lane >> 4;
  const int m_base  = (w & 3) * 16;
  const int n_base0 = (w >> 2) * 16;   // 0 or 16
  const int n_base1 = n_base0 + 32;    // 32 or 48

  v8f c0a = {}, c0b = {}, c1a = {}, c1b = {};

  for (int k0 = 0; k0 < kE; k0 += 64) {
    __syncthreads();
#pragma unroll
    for (int i = 0; i < 4; ++i) {          // A tile via async DMA (gather)
      int f = tid + i * 256;
      int row = f >> 4, kq = (f & 15) * 4;
      cp_async16(&As[row][kq], &emb[(size_t)toks[row] * kE + k0 + kq]);
    }
#pragma unroll
    for (int i = 0; i < 4; ++i) {          // B tile (transpose -> paired)
      int f = tid + i * 256;
      int n = f >> 4, kq = (f & 15) * 4;
      float4 v = *(const float4*)&Wih[(size_t)(N0 + n) * kE + k0 + kq];
      Bs[(kq >> 1) + 0][2 * n + 0] = v.x;
      Bs[(kq >> 1) + 0][2 * n + 1] = v.y;
      Bs[(kq >> 1) + 1][2 * n + 0] = v.z;
      Bs[(kq >> 1) + 1][2 * n + 1] = v.w;
    }
    cp_async_wait();
    __syncthreads();
#pragma unroll
    for (int kk = 0; kk < 64; kk += 4) {
      int koff = kk + hi * 2;
      int k2 = koff >> 1;
      float2 av = *(const float2*)&As[m_base + ml][koff];
      v2f a; a.x = av.x; a.y = av.y;
      float2 bv0 = *(const float2*)&Bs[k2][2 * (n_base0 + ml)];
      float2 bv1 = *(const float2*)&Bs[k2][2 * (n_base1 + ml)];
      v2f b0; b0.x = bv0.x; b0.y = bv0.y;
      v2f b1; b1.x = bv1.x; b1.y = bv1.y;
      if ((kk >> 2) & 1) { c0b = WMMA_F32(a, b0, c0b); c1b = WMMA_F32(a, b1, c1b); }
      else               { c0a = WMMA_F32(a, b0, c0a); c1a = WMMA_F32(a, b1, c1a); }
    }
  }
  v8f c0 = c0a + c0b, c1 = c1a + c1b;

  const int n0g = N0 + n_base0 + ml;
  const int n1g = N0 + n_base1 + ml;
  const float bias0 = bih[n0g] + bhh[n0g];
  const float bias1 = bih[n1g] + bhh[n1g];
#pragma unroll
  for (int r = 0; r < 8; ++r) {
    int m = M0 + m_base + hi * 8 + r;
    XG[(size_t)m * kH + n0g] = c0[r] + bias0;
    XG[(size_t)m * kH + n1g] = c1[r] + bias1;
  }
}

// ---------------------------------------------------------------------------
__global__ void zero_counter(unsigned* c) { *c = 0u; }

// ---------------------------------------------------------------------------
// Kernel 2: persistent RNN scan.  32 blocks x 128 threads (4 waves).
// Whh^T slice cached (paired) in LDS for all 512 steps; h chunk staged by
// async DMA; 4 independent WMMA accumulator chains hide WMMA RAW latency.
// ---------------------------------------------------------------------------
__global__ __launch_bounds__(128) void rnn_scan(
    const float* __restrict__ XG, const float* __restrict__ Whh,
    float* __restrict__ HS, float* __restrict__ hbuf,
    unsigned* __restrict__ counter)
{
  __shared__ float Bs[kH / 2][32];   // paired Whh^T: [k/2][2n + (k&1)]
  __shared__ float As[64][68];       // h chunk: [row][k]

  const int tid = threadIdx.x;
  const int N0 = blockIdx.x * 16;
  const int w = tid >> 5, lane = tid & 31;
  const int ml = lane & 15, hi = lane >> 4;
  const int m_base = w * 16;
  const int ng = N0 + ml;

  for (int i = 0; i < 64; ++i) {            // preload Whh^T slice (one time)
    int idx = tid + i * 128;                // 0..8191
    int k = idx >> 4, n = idx & 15;
    Bs[k >> 1][2 * n + (k & 1)] = Whh[(size_t)(N0 + n) * kH + k];
  }
  __syncthreads();

  const unsigned nblocks = gridDim.x;
  float* h0 = hbuf;
  float* h1 = hbuf + (size_t)kB * kH;

  for (int t = 0; t < kT; ++t) {
    float* h_next = (t & 1) ? h0 : h1;
    const float* h_prev = (t & 1) ? h1 : h0;

    v8f acc0 = {}, acc1 = {}, acc2 = {}, acc3 = {};
    if (t > 0) {
      for (int k0 = 0; k0 < kH; k0 += 64) {
#pragma unroll
        for (int i = 0; i < 8; ++i) {       // stage h chunk via async DMA
          int f = tid + i * 128;
          int row = f >> 4, kq = (f & 15) * 4;
          cp_async16(&As[row][kq], &h_prev[(size_t)row * kH + k0 + kq]);
        }
        cp_async_wait();
        __syncthreads();
#pragma unroll
        for (int kk = 0; kk < 64; kk += 4) {
          int koff = kk + hi * 2;
          float2 av = *(const float2*)&As[m_base + ml][koff];
          v2f a; a.x = av.x; a.y = av.y;
          float2 bv = *(const float2*)&Bs[(k0 + koff) >> 1][2 * ml];
          v2f b; b.x = bv.x; b.y = bv.y;
          switch ((kk >> 2) & 3) {           // 4 independent chains
            case 0: acc0 = WMMA_F32(a, b, acc0); break;
            case 1: acc1 = WMMA_F32(a, b, acc1); break;
            case 2: acc2 = WMMA_F32(a, b, acc2); break;
            default: acc3 = WMMA_F32(a, b, acc3); break;
          }
        }
        __syncthreads();
      }
    }
    v8f c = (acc0 + acc1) + (acc2 + acc3);

    // h_new = tanh(c + g); write state + sequence output
#pragma unroll
    for (int r = 0; r < 8; ++r) {
      int bm = m_base + hi * 8 + r;
      size_t gidx = ((size_t)bm * kT + t) * kH + ng;
      float hv = tanhf(c[r] + XG[gidx]);
      h_next[(size_t)bm * kH + ng] = hv;
      HS[gidx] = hv;
    }
    // device-wide barrier (arrive + spin on monotonically increasing counter)
    __syncthreads();
    if (tid == 0) {
      __threadfence();
      __hip_atomic_fetch_add(counter, 1u, __ATOMIC_RELEASE, __HIP_MEMORY_SCOPE_AGENT);
      unsigned target = nblocks * (unsigned)(t + 1);
      while (__hip_atomic_load(counter, __ATOMIC_ACQUIRE, __HIP_MEMORY_SCOPE_AGENT) < target)
        __builtin_amdgcn_s_sleep(1);
    }
    __syncthreads();
  }
}

// ---------------------------------------------------------------------------
// Kernel 3: fused head.  hidden = relu(HS @ W1^T + b1); out = hidden @ W2^T + b2
// ---------------------------------------------------------------------------
__global__ __launch_bounds__(256) void head_kernel(
    const float* __restrict__ HS, const float* __restrict__ W1,
    const float* __restrict__ b1, const float* __restrict__ W2,
    const float* __restrict__ b2, float* __restrict__ out)
{
  __shared__ float As[64][68];
  __shared__ float Bs[32][160];  // paired W1^T
  __shared__ float Hs[64][68];   // relu(hidden) tile
  __shared__ float W2s[64][36];  // W2^T zero-padded to 32 cols

  const int tid = threadIdx.x;
  const int M0 = blockIdx.x * 64;
  const int w = tid >> 5, lane = tid & 31;
  const int ml = lane & 15, hi = lane >> 4;
  const int m_base  = (w & 3) * 16;
  const int n_base0 = (w >> 2) * 16;
  const int n_base1 = n_base0 + 32;

#pragma unroll
  for (int i = 0; i < 8; ++i) {            // W2^T (padded) -> LDS
    int e = tid + i * 256;
    int cc = e >> 5, oo = e & 31;
    W2s[cc][oo] = (oo < kO) ? W2[(size_t)oo * kC + cc] : 0.f;
  }

  v8f c0a = {}, c0b = {}, c1a = {}, c1b = {};
  for (int k0 = 0; k0 < kH; k0 += 64) {    // fc1, K = 512
    __syncthreads();
#pragma unroll
    for (int i = 0; i < 4; ++i) {          // A tile via async DMA
      int f = tid + i * 256;
      int row = f >> 4, kq = (f & 15) * 4;
      cp_async16(&As[row][kq], &HS[(size_t)(M0 + row) * kH + k0 + kq]);
    }
#pragma unroll
    for (int i = 0; i < 4; ++i) {          // W1^T tile (transpose -> paired)
      int f = tid + i * 256;
      int n = f >> 4, kq = (f & 15) * 4;
      float4 v = *(const float4*)&W1[(size_t)n * kH + k0 + kq];
      Bs[(kq >> 1) + 0][2 * n + 0] = v.x;
      Bs[(kq >> 1) + 0][2 * n + 1] = v.y;
      Bs[(kq >> 1) + 1][2 * n + 0] = v.z;
      Bs[(kq >> 1) + 1][2 * n + 1] = v.w;
    }
    cp_async_wait();
    __syncthreads();
#pragma unroll
    for (int kk = 0; kk < 64; kk += 4) {
      int koff = kk + hi * 2;
      int k2 = koff >> 1;
      float2 av = *(const float2*)&As[m_base + ml][koff];
      v2f a; a.x = av.x; a.y = av.y;
      float2 bv0 = *(const float2*)&Bs[k2][2 * (n_base0 + ml)];
      float2 bv1 = *(const float2*)&Bs[k2][2 * (n_base1 + ml)];
      v2f b0; b0.x = bv0.x; b0.y = bv0.y;
      v2f b1; b1.x = bv1.x; b1.y = bv1.y;
      if ((kk >> 2) & 1) { c0b = WMMA_F32(a, b0, c0b); c1b = WMMA_F32(a, b1, c1b); }
      else               { c0a = WMMA_F32(a, b0, c0a); c1a = WMMA_F32(a, b1, c1a); }
    }
  }
  v8f c0 = c0a + c0b, c1 = c1a + c1b;
  __syncthreads();
  {                                         // relu + b1 -> LDS hidden tile
    float bb0 = b1[n_base0 + ml];
    float bb1 = b1[n_base1 + ml];
#pragma unroll
    for (int r = 0; r < 8; ++r) {
      int row = m_base + hi * 8 + r;
      float h0v = c0[r] + bb0; Hs[row][n_base0 + ml] = h0v > 0.f ? h0v : 0.f;
      float h1v = c1[r] + bb1; Hs[row][n_base1 + ml] = h1v > 0.f ? h1v : 0.f;
    }
  }
  __syncthreads();
  // fc2: 64 x 32(padded), K = 64.  8 tiles, one per wave, 2 chains.
  const int o_base = (w >> 2) * 16;
  v8f c2a = {}, c2b = {};
#pragma unroll
  for (int kk = 0; kk < 64; kk += 4) {
    int koff = kk + hi * 2;
    float2 av = *(const float2*)&Hs[m_base + ml][koff];
    v2f a; a.x = av.x; a.y = av.y;
    v2f b; b.x = W2s[koff][o_base + ml]; b.y = W2s[koff + 1][o_base + ml];
    if ((kk >> 2) & 1) c2b = WMMA_F32(a, b, c2b);
    else               c2a = WMMA_F32(a, b, c2a);
  }
  v8f c2 = c2a + c2b;
  const int og = o_base + ml;
  if (og < kO) {
    float bb = b2[og];
#pragma unroll
    for (int r = 0; r < 8; ++r) {
      int m = M0 + m_base + hi * 8 + r;
      out[(size_t)m * kO + og] = c2[r] + bb;
    }
  }
}

// ---------------------------------------------------------------------------
extern "C" void kernel_launch(void* const* d_in, const int* in_sizes, int n_in,
                              void* d_out, int out_size, void* d_ws, size_t ws_size,
                              hipStream_t stream)
{
  const int*   x    = (const int*)d_in[0];
  const float* embt = (const float*)d_in[1];
  const float* Wih  = (const float*)d_in[2];
  const float* Whh  = (const float*)d_in[3];
  const float* bih  = (const float*)d_in[4];
  const float* bhh  = (const float*)d_in[5];
  const float* W1   = (const float*)d_in[6];
  const float* b1   = (const float*)d_in[7];
  const float* W2   = (const float*)d_in[8];
  const float* b2   = (const float*)d_in[9];

  char* ws = (char*)d_ws;
  float* XG   = (float*)ws;                                       // 64 MB
  float* HS   = (float*)(ws + (size_t)kM * kH * 4);               // 64 MB
  float* hbuf = (float*)(ws + (size_t)2 * kM * kH * 4);           // 256 KB
  unsigned* counter =
      (unsigned*)(ws + (size_t)2 * kM * kH * 4 + (size_t)2 * kB * kH * 4);

  xg_kernel<<<dim3(kM / 64, kH / 64), 256, 0, stream>>>(x, embt, Wih, bih, bhh, XG);
  zero_counter<<<1, 1, 0, stream>>>(counter);
  rnn_scan<<<32, 128, 0, stream>>>(XG, Whh, HS, hbuf, counter);
  head_kernel<<<kM / 64, 256, 0, stream>>>(HS, W1, b1, W2, b2, (float*)d_out);
}